// GraphSAGERiskModel_79920751444079
// MI455X (gfx1250) — compile-verified
//
#include <hip/hip_runtime.h>
#include <hip/hip_bf16.h>

typedef __attribute__((ext_vector_type(2))) float v2f;
typedef __attribute__((ext_vector_type(8))) float v8f;

#define WG 256

// ---------------------------------------------------------------- utilities
__global__ __launch_bounds__(WG) void zero_kernel(float* __restrict__ p, long long n) {
    long long i = (long long)blockIdx.x * WG + threadIdx.x;
    long long stride = (long long)gridDim.x * WG;
    for (; i < n; i += stride) p[i] = 0.0f;
}

__global__ __launch_bounds__(WG) void degree_kernel(const int* __restrict__ dst,
                                                    float* __restrict__ deg, int nE) {
    int i = blockIdx.x * WG + threadIdx.x;
    int stride = gridDim.x * WG;
    for (; i < nE; i += stride)
        __hip_atomic_fetch_add(&deg[dst[i]], 1.0f, __ATOMIC_RELAXED, __HIP_MEMORY_SCOPE_AGENT);
}

__global__ __launch_bounds__(WG) void deginv_kernel(float* __restrict__ deg, int n) {
    int i = blockIdx.x * WG + threadIdx.x;
    if (i < n) deg[i] = 1.0f / fmaxf(deg[i], 1.0f);
}

// ------------------------------------------------- edge-parallel aggregation
// One thread handles one (edge, 4-feature chunk): float4 gather from feat[src],
// four native f32 atomic adds into agg[dst]. Node features stay L2-resident.
template <int F>
__global__ __launch_bounds__(WG) void aggregate_kernel(const float* __restrict__ feat,
                                                       const int* __restrict__ src,
                                                       const int* __restrict__ dst,
                                                       float* __restrict__ agg, int nE) {
    constexpr int C = F / 4;
    long long total = (long long)nE * C;
    long long i = (long long)blockIdx.x * WG + threadIdx.x;
    long long stride = (long long)gridDim.x * WG;
    for (; i < total; i += stride) {
        int e = (int)(i / C);
        int c = (int)(i % C);
        int s = src[e];
        int d = dst[e];
        const float4 v = *(const float4*)(feat + (size_t)s * F + c * 4);
        float* p = agg + (size_t)d * F + c * 4;
        __hip_atomic_fetch_add(p + 0, v.x, __ATOMIC_RELAXED, __HIP_MEMORY_SCOPE_AGENT);
        __hip_atomic_fetch_add(p + 1, v.y, __ATOMIC_RELAXED, __HIP_MEMORY_SCOPE_AGENT);
        __hip_atomic_fetch_add(p + 2, v.z, __ATOMIC_RELAXED, __HIP_MEMORY_SCOPE_AGENT);
        __hip_atomic_fetch_add(p + 3, v.w, __ATOMIC_RELAXED, __HIP_MEMORY_SCOPE_AGENT);
    }
}

// ------------------------------------------------------ WMMA dense SAGE layer
// out[N,FOUT] = act( bn( (agg*dinv) @ wl + xin @ wr + bias ) )
// One wave per 16-row tile, V_WMMA_F32_16X16X4_F32.
// Weights staged in LDS *pre-swizzled* so each lane's B fragment
// {W[k][n], W[k+1][n]} is one aligned ds_load_b64 into an even VGPR pair.
template <int FIN, int FOUT, bool BN>
__global__ __launch_bounds__(WG) void sage_dense_kernel(
    const float* __restrict__ agg, const float* __restrict__ xin,
    const float* __restrict__ dinv,
    const float* __restrict__ wl, const float* __restrict__ wr,
    const float* __restrict__ bias,
    const float* __restrict__ bn_g, const float* __restrict__ bn_b,
    const float* __restrict__ bn_m, const float* __restrict__ bn_v,
    float* __restrict__ out, int nrows) {
    __shared__ v2f lwl[(FIN / 2) * FOUT];  // lwl[(k>>1)*FOUT + n] = {W[k][n], W[k+1][n]}
    __shared__ v2f lwr[(FIN / 2) * FOUT];
    {
        float* fl = (float*)lwl;
        float* fr = (float*)lwr;
        for (int i = threadIdx.x; i < FIN * FOUT; i += WG) {
            const int k = i / FOUT;
            const int n = i - k * FOUT;
            const int sw = ((k >> 1) * FOUT + n) * 2 + (k & 1);
            fl[sw] = wl[i];
            fr[sw] = wr[i];
        }
    }
    __syncthreads();

    int ntiles = (nrows + 15) >> 4;
    int tile = blockIdx.x * (WG / 32) + (threadIdx.x >> 5);
    if (tile >= ntiles) return;  // wave-uniform: EXEC stays all-ones for WMMA

    const int lane = threadIdx.x & 31;
    const int half = lane >> 4;   // 0: K={0,1}, rows M..M+7 ; 1: K={2,3}, rows M+8..
    const int ln = lane & 15;
    const int row0 = tile * 16;
    int m = row0 + ln;
    if (m >= nrows) m = nrows - 1;  // clamped load (dup); stores are guarded
    const float dm = dinv[m];

    constexpr int NC = FOUT / 16;
    v8f acc[NC];
#pragma unroll
    for (int ct = 0; ct < NC; ++ct) {
        float bv = bias[ct * 16 + ln];
        acc[ct] = v8f{bv, bv, bv, bv, bv, bv, bv, bv};
    }

    const float* arow = agg + (size_t)m * FIN;
    const float* xrow = xin + (size_t)m * FIN;
#pragma unroll
    for (int k0 = 0; k0 < FIN; k0 += 4) {
        const int ka = k0 + half * 2;  // per-lane K pair per ISA 16x4 f32 A layout
        v2f aA = *(const v2f*)(arow + ka);  // contiguous -> global_load_b64
        v2f aX = *(const v2f*)(xrow + ka);
        aA *= dm;                            // fold deg_inv (v_pk_mul_f32)
        const int kp = ka >> 1;
#pragma unroll
        for (int ct = 0; ct < NC; ++ct) {
            const int n = ct * 16 + ln;
            v2f bL = lwl[kp * FOUT + n];  // one ds_load_b64, even-aligned pair
            acc[ct] = __builtin_amdgcn_wmma_f32_16x16x4_f32(
                false, aA, false, bL, (short)0, acc[ct], false, false);
            v2f bR = lwr[kp * FOUT + n];
            acc[ct] = __builtin_amdgcn_wmma_f32_16x16x4_f32(
                false, aX, false, bR, (short)0, acc[ct], false, false);
        }
    }

    // epilogue: BN(eval) folded to scale/shift, ReLU, scatter per C/D layout
#pragma unroll
    for (int ct = 0; ct < NC; ++ct) {
        const int n = ct * 16 + ln;
        float scale = 1.0f, shift = 0.0f;
        if (BN) {
            const float sc = bn_g[n] * rsqrtf(bn_v[n] + 1e-5f);
            scale = sc;
            shift = bn_b[n] - bn_m[n] * sc;
        }
#pragma unroll
        for (int j = 0; j < 8; ++j) {
            const int row = row0 + half * 8 + j;  // VGPR j holds M=j (lanes<16) / M=j+8
            if (row < nrows) {
                float v = acc[ct][j] * scale + shift;
                out[(size_t)row * FOUT + n] = fmaxf(v, 0.0f);
            }
        }
    }
}

// --------------------------------------------------------------- fc + sigmoid
__global__ __launch_bounds__(WG) void fc_sigmoid_kernel(const float* __restrict__ h,
                                                        const float* __restrict__ fw,
                                                        const float* __restrict__ fb,
                                                        float* __restrict__ out, int n) {
    int i = blockIdx.x * WG + threadIdx.x;
    if (i >= n) return;
    const float* r = h + (size_t)i * 32;
    float s = fb[0];
#pragma unroll
    for (int j = 0; j < 32; ++j) s += r[j] * fw[j];
    out[i] = 1.0f / (1.0f + expf(-s));
}

// ------------------------------------------------------------------- launcher
extern "C" void kernel_launch(void* const* d_in, const int* in_sizes, int n_in,
                              void* d_out, int out_size, void* d_ws, size_t ws_size,
                              hipStream_t stream) {
    const float* x = (const float*)d_in[0];
    const int* ei = (const int*)d_in[1];
    const float* w1l = (const float*)d_in[2];
    const float* w1r = (const float*)d_in[3];
    const float* b1 = (const float*)d_in[4];
    const float* bn1_g = (const float*)d_in[5];
    const float* bn1_b = (const float*)d_in[6];
    const float* bn1_m = (const float*)d_in[7];
    const float* bn1_v = (const float*)d_in[8];
    const float* w2l = (const float*)d_in[9];
    const float* w2r = (const float*)d_in[10];
    const float* b2 = (const float*)d_in[11];
    const float* bn2_g = (const float*)d_in[12];
    const float* bn2_b = (const float*)d_in[13];
    const float* bn2_m = (const float*)d_in[14];
    const float* bn2_v = (const float*)d_in[15];
    const float* w3l = (const float*)d_in[16];
    const float* w3r = (const float*)d_in[17];
    const float* b3 = (const float*)d_in[18];
    const float* fc_w = (const float*)d_in[19];
    const float* fc_b = (const float*)d_in[20];

    const int N = in_sizes[0] / 8;
    const int E = in_sizes[1] / 2;
    const int* src = ei;
    const int* dst = ei + E;

    float* ws = (float*)d_ws;
    float* dinv = ws;                    // N
    float* aggb = dinv + N;              // 64N
    float* hA = aggb + (size_t)64 * N;   // 64N
    float* hB = hA + (size_t)64 * N;     // 64N

    const int nblk = (N + WG - 1) / WG;
    const int eblk = (E + WG - 1) / WG;
    const int ntiles = (N + 15) / 16;
    const int dblk = (ntiles + (WG / 32) - 1) / (WG / 32);
    auto agg_blocks = [](long long total) {
        long long b = (total + WG - 1) / WG;
        if (b > 262144) b = 262144;
        return (int)b;
    };

    // degree -> 1/max(deg,1)
    zero_kernel<<<nblk, WG, 0, stream>>>(dinv, N);
    degree_kernel<<<eblk, WG, 0, stream>>>(dst, dinv, E);
    deginv_kernel<<<nblk, WG, 0, stream>>>(dinv, N);

    // layer 1: 8 -> 64, BN + ReLU
    zero_kernel<<<agg_blocks((long long)N * 8), WG, 0, stream>>>(aggb, (long long)N * 8);
    aggregate_kernel<8><<<agg_blocks((long long)E * 2), WG, 0, stream>>>(x, src, dst, aggb, E);
    sage_dense_kernel<8, 64, true><<<dblk, WG, 0, stream>>>(
        aggb, x, dinv, w1l, w1r, b1, bn1_g, bn1_b, bn1_m, bn1_v, hA, N);

    // layer 2: 64 -> 64, BN + ReLU
    zero_kernel<<<agg_blocks((long long)N * 64), WG, 0, stream>>>(aggb, (long long)N * 64);
    aggregate_kernel<64><<<agg_blocks((long long)E * 16), WG, 0, stream>>>(hA, src, dst, aggb, E);
    sage_dense_kernel<64, 64, true><<<dblk, WG, 0, stream>>>(
        aggb, hA, dinv, w2l, w2r, b2, bn2_g, bn2_b, bn2_m, bn2_v, hB, N);

    // layer 3: 64 -> 32, ReLU
    zero_kernel<<<agg_blocks((long long)N * 64), WG, 0, stream>>>(aggb, (long long)N * 64);
    aggregate_kernel<64><<<agg_blocks((long long)E * 16), WG, 0, stream>>>(hB, src, dst, aggb, E);
    sage_dense_kernel<64, 32, false><<<dblk, WG, 0, stream>>>(
        aggb, hB, dinv, w3l, w3r, b3, nullptr, nullptr, nullptr, nullptr, hA, N);

    // fc + sigmoid
    fc_sigmoid_kernel<<<nblk, WG, 0, stream>>>(hA, fc_w, fc_b, (float*)d_out, N);
}